// pWasserstein_66829691126449
// MI455X (gfx1250) — compile-verified
//
#include <hip/hip_runtime.h>
#include <math.h>

// p-Wasserstein (P=2) between two 1-D histograms, L = 2^23.
// Pipeline: sums -> scans (CDFs) -> inverse-CDF sampling -> transport interp -> weighted mean.
// Bandwidth-bound (~400MB HBM traffic, ~17us floor @ 23.3 TB/s); J0/J1/XI0/XI1 (128MB) are
// L2-resident (192MB), so binary searches stay on-chip. fp32 precision required -> use
// CDNA5's fp32 matrix op V_WMMA_F32_16X16X4_F32 for exact 16x16 tile prefix sums (L*X with
// triangular-ones A operand, chained over K in 4-wide slices). The WMMA scan kernel is
// branch-free (full chunks only, float4/b128 staging); any tail is handled by a separate
// scalar kernel launched only when L % CHUNK != 0 (launch set is a pure function of L).

#define EPS 1e-7f

typedef __attribute__((ext_vector_type(2))) float v2f;
typedef __attribute__((ext_vector_type(8))) float v8f;

// ---------------- pass 1: per-block sums of (x + EPS) ----------------
__global__ __launch_bounds__(256) void block_sum_kernel(const float* __restrict__ in,
                                                        float* __restrict__ bs,
                                                        int L, int chunk) {
  __shared__ float red[256];
  long base = (long)blockIdx.x * chunk;
  float s = 0.f;
  if (base + chunk <= (long)L) {  // full chunk: vectorized, unguarded
    const float4* in4 = reinterpret_cast<const float4*>(in + base);
    for (int i = threadIdx.x; i < chunk / 4; i += 256) {
      float4 v = in4[i];
      s += (v.x + EPS) + (v.y + EPS) + (v.z + EPS) + (v.w + EPS);
    }
  } else {
    for (int i = threadIdx.x; i < chunk; i += 256) {
      long g = base + i;
      s += (g < L) ? (in[g] + EPS) : 0.f;
    }
  }
  red[threadIdx.x] = s;
  __syncthreads();
  for (int off = 128; off > 0; off >>= 1) {
    if (threadIdx.x < off) red[threadIdx.x] += red[threadIdx.x + off];
    __syncthreads();
  }
  if (threadIdx.x == 0) bs[blockIdx.x] = red[0];
}

// ------- pass 2: exclusive scan of block sums (in place) + total at bs[nb] -------
__global__ __launch_bounds__(1024) void offsets_kernel(float* __restrict__ bs, int nb) {
  if (nb > 1024) {  // generic fallback (not hit for L = 2^23)
    if (threadIdx.x == 0) {
      float run = 0.f;
      for (int i = 0; i < nb; ++i) { float v = bs[i]; bs[i] = run; run += v; }
      bs[nb] = run;
    }
    return;
  }
  __shared__ float sh[1024];
  int t = threadIdx.x;
  float v = (t < nb) ? bs[t] : 0.f;
  sh[t] = v;
  __syncthreads();
  for (int d = 1; d < 1024; d <<= 1) {
    float x = (t >= d) ? sh[t - d] : 0.f;
    __syncthreads();
    sh[t] += x;
    __syncthreads();
  }
  float incl = sh[t];
  if (t < nb) bs[t] = incl - v;      // exclusive offset
  if (t == nb - 1) bs[nb] = incl;    // grand total
}

// ------- pass 3: WMMA tile scan -> normalized CDF J = cumsum(x+EPS)/total -------
// FULL CHUNKS ONLY (branch-free). Tile layout: 256 elements per wave, column-major
// X[m,n] = e(m+16n). Y = L*X (L = lower-triangular ones) gives within-column inclusive
// scans via 4 chained 16x16x4 fp32 WMMAs; column/cross-wave offsets via shuffles + LDS.
__global__ __launch_bounds__(256) void scan_wmma_kernel(const float* __restrict__ in,
                                                        const float* __restrict__ bs,
                                                        float* __restrict__ J,
                                                        int chunk, int nb) {
  __shared__ float stage[2048];
  __shared__ float wtot[8];
  const int tid = threadIdx.x;
  const int lane = tid & 31;
  const int wid = tid >> 5;
  const int col = lane & 15;                  // column owned by this lane
  const int koff = (lane < 16) ? 0 : 2;       // K offset within 4-wide slice (A/B layout)
  const int mhalf = (lane < 16) ? 0 : 8;      // D layout row half
  const long base = (long)blockIdx.x * chunk;
  const float blockOff = bs[blockIdx.x];
  const float invS = 1.0f / bs[nb];
  float carry = 0.f;
  const int iters = chunk / 2048;

  for (int it = 0; it < iters; ++it) {
    __syncthreads();  // protect stage/wtot from previous iteration's readers
    const long ibase = base + (long)it * 2048;
    {  // vectorized staging: 512 float4 per iteration (global_load_b128 -> ds_store_b128)
      const float4* in4 = reinterpret_cast<const float4*>(in + ibase);
      float4* st4 = reinterpret_cast<float4*>(stage);
#pragma unroll
      for (int r = 0; r < 2; ++r) {
        float4 v = in4[tid + 256 * r];
        v.x += EPS; v.y += EPS; v.z += EPS; v.w += EPS;
        st4[tid + 256 * r] = v;
      }
    }
    __syncthreads();

    const float* tile = &stage[wid * 256];
    v8f acc = {};
#pragma unroll
    for (int j = 0; j < 4; ++j) {
      const int k0 = 4 * j + koff;
      // A_j = L[:,4j:4j+4], L[m,k] = (m >= k); A layout: lane<16 -> K=4j|4j+1, lane>=16 -> K=4j+2|4j+3
      v2f a;
      a.x = (col >= k0) ? 1.f : 0.f;
      a.y = (col >= k0 + 1) ? 1.f : 0.f;
      // B_j = X[4j:4j+4,:]; B layout mirrors: VGPR0 -> K=k0, VGPR1 -> K=k0+1, N = col
      v2f b;
      b.x = tile[16 * col + k0];
      b.y = tile[16 * col + k0 + 1];
      acc = __builtin_amdgcn_wmma_f32_16x16x4_f32(false, a, false, b, (short)0, acc,
                                                  false, false);
    }
    // Column totals: Y[15,col] lives in VGPR7 of lane col+16 (D layout).
    float csum = __shfl(acc[7], col + 16, 32);
    // Inclusive shuffle-scan of the 16 column sums (both lane halves mirror).
    float s = csum;
#pragma unroll
    for (int d = 1; d < 16; d <<= 1) {
      float t = __shfl(s, (col - d) & 31, 32);
      if (col >= d) s += t;
    }
    const float ecs = s - csum;                 // exclusive column prefix
    const float wave_total = __shfl(s, 15, 32); // total of this 256-tile
    if (lane == 0) wtot[wid] = wave_total;
    __syncthreads();
    float woff = carry, allt = 0.f;
    for (int w = 0; w < 8; ++w) {
      float v = wtot[w];
      if (w < wid) woff += v;
      allt += v;
    }
    // Scatter scanned values back to stage in flatten order e = m + 16n.
#pragma unroll
    for (int r = 0; r < 8; ++r) {
      int m = mhalf + r;
      stage[wid * 256 + m + 16 * col] = acc[r] + ecs + woff;
    }
    carry += allt;
    __syncthreads();
    {  // vectorized normalized-CDF writeback (ds_load_b128 -> global_store_b128)
      float4* J4 = reinterpret_cast<float4*>(J + ibase);
      const float4* st4 = reinterpret_cast<const float4*>(stage);
#pragma unroll
      for (int r = 0; r < 2; ++r) {
        float4 v = st4[tid + 256 * r];
        float4 o;
        o.x = (v.x + blockOff) * invS;
        o.y = (v.y + blockOff) * invS;
        o.z = (v.z + blockOff) * invS;
        o.w = (v.w + blockOff) * invS;
        J4[tid + 256 * r] = o;
      }
    }
  }
}

// ------- tail scan (single block): only launched when L % CHUNK != 0 -------
__global__ __launch_bounds__(256) void scan_tail_kernel(const float* __restrict__ in,
                                                        const float* __restrict__ bs,
                                                        float* __restrict__ J,
                                                        int L, long start, int bidx, int nb) {
  __shared__ float sh[256];
  float carry = bs[bidx];              // exclusive offset of the tail block
  const float invS = 1.0f / bs[nb];
  for (long ib = start; ib < L; ib += 256) {
    long g = ib + threadIdx.x;
    float v = (g < L) ? (in[g] + EPS) : 0.f;
    sh[threadIdx.x] = v;
    __syncthreads();
    for (int d = 1; d < 256; d <<= 1) {
      float x = (threadIdx.x >= d) ? sh[threadIdx.x - d] : 0.f;
      __syncthreads();
      sh[threadIdx.x] += x;
      __syncthreads();
    }
    float incl = sh[threadIdx.x];
    if (g < L) J[g] = (incl + carry) * invS;
    float tot = sh[255];
    __syncthreads();
    carry += tot;
  }
}

// ------- inverse CDF: XI[k] = interp(k/(L-1), J, arange), XI[0] forced 0 -------
__global__ __launch_bounds__(256) void invcdf_kernel(const float* __restrict__ Jc,
                                                     float* __restrict__ XI,
                                                     int L, float step) {
  long k = (long)blockIdx.x * 256 + threadIdx.x;
  if (k >= L) return;
  float xt = (float)k * step;
  int lo = 0, hi = L;                 // searchsorted side='right'
  while (lo < hi) {
    int mid = (lo + hi) >> 1;
    if (Jc[mid] <= xt) lo = mid + 1; else hi = mid;
  }
  int idx = lo - 1;
  idx = idx < 0 ? 0 : idx;
  idx = idx > L - 2 ? L - 2 : idx;
  float st = Jc[idx], ed = Jc[idx + 1];
  float w = (xt - st) / (ed - st);
  float res = (float)idx + w;         // fp = arange -> lerp is idx + w
  if (k == 0) res = 0.f;
  XI[k] = res;
}

// ------- transport displacement + weighted P=2 partial reduction -------
__global__ __launch_bounds__(256) void transport_kernel(const float* __restrict__ XI0,
                                                        const float* __restrict__ XI1,
                                                        const float* __restrict__ inp,
                                                        const float* __restrict__ S0ptr,
                                                        float* __restrict__ partials, int L) {
  __shared__ float red[256];
  const float invS = 1.0f / (*S0ptr);
  float acc = 0.f;
  for (long i = (long)blockIdx.x * 256 + threadIdx.x; i < L; i += (long)gridDim.x * 256) {
    float q = (float)i;
    int lo = 0, hi = L;               // searchsorted(XI0, q, 'right')
    while (lo < hi) {
      int mid = (lo + hi) >> 1;
      if (XI0[mid] <= q) lo = mid + 1; else hi = mid;
    }
    int idx = lo - 1;
    idx = idx < 0 ? 0 : idx;
    idx = idx > L - 2 ? L - 2 : idx;
    float st = XI0[idx], ed = XI0[idx + 1];
    float f0 = st - XI1[idx];
    float f1 = ed - XI1[idx + 1];
    float w = (q - st) / (ed - st);
    float u = f0 + w * (f1 - f0);
    if (i == 0) u = 0.f;              // _interp forces result[0] = 0
    float I0 = (inp[i] + EPS) * invS;
    acc += u * u * I0;                // |u|^P * I0, P = 2
  }
  red[threadIdx.x] = acc;
  __syncthreads();
  for (int off = 128; off > 0; off >>= 1) {
    if (threadIdx.x < off) red[threadIdx.x] += red[threadIdx.x + off];
    __syncthreads();
  }
  if (threadIdx.x == 0) partials[blockIdx.x] = red[0];
}

__global__ __launch_bounds__(1024) void final_kernel(const float* __restrict__ partials,
                                                     int n, float* __restrict__ out,
                                                     float invL) {
  __shared__ float red[1024];
  int t = threadIdx.x;
  float s = 0.f;
  for (int i = t; i < n; i += 1024) s += partials[i];
  red[t] = s;
  __syncthreads();
  for (int off = 512; off > 0; off >>= 1) {
    if (t < off) red[t] += red[t + off];
    __syncthreads();
  }
  if (t == 0) out[0] = sqrtf(red[0] * invL);   // (mean)^(1/P), P = 2
}

extern "C" void kernel_launch(void* const* d_in, const int* in_sizes, int n_in,
                              void* d_out, int out_size, void* d_ws, size_t ws_size,
                              hipStream_t stream) {
  const float* in0 = (const float*)d_in[0];   // inputs
  const float* in1 = (const float*)d_in[1];   // target
  float* out = (float*)d_out;
  const int L = in_sizes[0];
  const int CHUNK = 8192;
  const int NBfull = L / CHUNK;               // 1024 for L = 2^23
  const int NB = (L + CHUNK - 1) / CHUNK;
  const int NPART = 1024;

  // Workspace layout (floats): J0|J1|XI0|XI1|bs0[NB+1]|bs1[NB+1]|partials[NPART]
  float* J0 = (float*)d_ws;
  float* J1 = J0 + (size_t)L;
  float* XI0 = J1 + (size_t)L;
  float* XI1 = XI0 + (size_t)L;
  float* bs0 = XI1 + (size_t)L;
  float* bs1 = bs0 + (NB + 1);
  float* partials = bs1 + (NB + 1);

  block_sum_kernel<<<NB, 256, 0, stream>>>(in0, bs0, L, CHUNK);
  block_sum_kernel<<<NB, 256, 0, stream>>>(in1, bs1, L, CHUNK);
  offsets_kernel<<<1, 1024, 0, stream>>>(bs0, NB);
  offsets_kernel<<<1, 1024, 0, stream>>>(bs1, NB);
  if (NBfull > 0) {
    scan_wmma_kernel<<<NBfull, 256, 0, stream>>>(in0, bs0, J0, CHUNK, NB);
    scan_wmma_kernel<<<NBfull, 256, 0, stream>>>(in1, bs1, J1, CHUNK, NB);
  }
  if (NB > NBfull) {  // tail exists only if L % CHUNK != 0 (pure function of L)
    const long start = (long)NBfull * CHUNK;
    scan_tail_kernel<<<1, 256, 0, stream>>>(in0, bs0, J0, L, start, NBfull, NB);
    scan_tail_kernel<<<1, 256, 0, stream>>>(in1, bs1, J1, L, start, NBfull, NB);
  }
  const int nb2 = (L + 255) / 256;
  const float step = 1.0f / (float)(L - 1);
  invcdf_kernel<<<nb2, 256, 0, stream>>>(J0, XI0, L, step);
  invcdf_kernel<<<nb2, 256, 0, stream>>>(J1, XI1, L, step);
  transport_kernel<<<NPART, 256, 0, stream>>>(XI0, XI1, in0, bs0 + NB, partials, L);
  final_kernel<<<1, 1024, 0, stream>>>(partials, NPART, out, 1.0f / (float)L);
}